// AdvancedTradingModel_87677462380895
// MI455X (gfx1250) — compile-verified
//
#include <hip/hip_runtime.h>
#include <hip/hip_bf16.h>

#define B_    16
#define T_    128
#define D_    128
#define H_    128
#define M_    512
#define EPS_  1e-8f
#define BIGC_ 50.0f
#define DTC_  0.1f
#define TOPK_ 8

typedef float v2f __attribute__((ext_vector_type(2)));
typedef float v8f __attribute__((ext_vector_type(8)));

// ---------------------------------------------------------------------------
// WMMA f32 16x16x4:  D = A(16x4) * B(4x16) + C(16x16), one matrix per wave32.
// ---------------------------------------------------------------------------
__device__ __forceinline__ v8f wmma4(v2f a, v2f b, v8f c) {
  // (neg_a, A, neg_b, B, c_mod, C, reuse_a, reuse_b)
  return __builtin_amdgcn_wmma_f32_16x16x4_f32(false, a, false, b, (short)0, c,
                                               false, false);
}

// A/B fragment loaders.
// A (16x4, MxK): lane<16 -> M=lane,   K=kk+{0,1}; lane>=16 -> M=lane-16, K=kk+{2,3}
// B (4x16, KxN): lane<16 -> N=lane,   K=kk+{0,1}; lane>=16 -> N=lane-16, K=kk+{2,3}
// ldfragT: element (r, k) at W[r*ld + k]  (covers A row-major and B = W^T)
__device__ __forceinline__ v2f ldfragT(const float* __restrict__ W, int ld,
                                       int r0, int kk, int lane) {
  const float* p = W + (size_t)(r0 + (lane & 15)) * ld + kk + ((lane >> 4) << 1);
  v2f f; f.x = p[0]; f.y = p[1]; return f;
}
// ldfragN: B[k][n] = W[k*ld + n] (non-transposed weight)
__device__ __forceinline__ v2f ldfragN(const float* __restrict__ W, int ld,
                                       int c0, int kk, int lane) {
  const float* p = W + (size_t)(kk + ((lane >> 4) << 1)) * ld + c0 + (lane & 15);
  v2f f; f.x = p[0]; f.y = p[ld]; return f;
}

__device__ __forceinline__ float sigmoidf_(float v) { return 1.0f / (1.0f + expf(-v)); }
__device__ __forceinline__ float safe_expf_(float v) { return expf(fminf(v, BIGC_)); }
__device__ __forceinline__ float softplusf_(float v) {
  return (v > 30.0f) ? v : log1pf(expf(v));
}

// ---------------------------------------------------------------------------
// Stage A: sensory = (x@sw^T+b)*sigmoid(x@smu^T+b)*safe_exp(x@ssig^T+b)
//          taux    = x@tc_x^T + tc_b          (rows = B*T = 2048, cols = H)
// ---------------------------------------------------------------------------
__global__ void stage_a_kernel(const float* __restrict__ x,
                               const float* __restrict__ sw_w,  const float* __restrict__ sw_b,
                               const float* __restrict__ smu_w, const float* __restrict__ smu_b,
                               const float* __restrict__ ssig_w,const float* __restrict__ ssig_b,
                               const float* __restrict__ tc_w,  const float* __restrict__ tc_b,
                               float* __restrict__ sensory, float* __restrict__ taux) {
  int lane = threadIdx.x & 31;
  int wave = threadIdx.x >> 5;
  int tile = blockIdx.x * (blockDim.x >> 5) + wave;   // 1024 tiles
  int row0 = (tile >> 3) * 16;                        // 128 row tiles
  int col0 = (tile & 7) * 16;                         // 8 col tiles
  v8f a0 = {}, a1 = {}, a2 = {}, a3 = {};
  for (int kk = 0; kk < D_; kk += 4) {
    v2f av = ldfragT(x, D_, row0, kk, lane);
    v2f b0 = ldfragT(sw_w,  D_, col0, kk, lane);
    v2f b1 = ldfragT(smu_w, D_, col0, kk, lane);
    v2f b2 = ldfragT(ssig_w,D_, col0, kk, lane);
    v2f b3 = ldfragT(tc_w, D_ + H_, col0, kk, lane);  // tc_x = tc_w[:, :D]
    a0 = wmma4(av, b0, a0);
    a1 = wmma4(av, b1, a1);
    a2 = wmma4(av, b2, a2);
    a3 = wmma4(av, b3, a3);
  }
  int n = col0 + (lane & 15);
#pragma unroll
  for (int v = 0; v < 8; ++v) {
    int m = row0 + v + ((lane >> 4) << 3);
    float s = (a0[v] + sw_b[n]) * sigmoidf_(a1[v] + smu_b[n]) * safe_expf_(a2[v] + ssig_b[n]);
    sensory[(size_t)m * H_ + n] = s;
    taux[(size_t)m * H_ + n]   = a3[v] + tc_b[n];
  }
}

// ---------------------------------------------------------------------------
// OUT = X @ W^T + bias   (nrows x 128)
// ---------------------------------------------------------------------------
__global__ void proj_nt_kernel(const float* __restrict__ X, const float* __restrict__ W,
                               const float* __restrict__ bias, float* __restrict__ OUT) {
  int lane = threadIdx.x & 31;
  int wave = threadIdx.x >> 5;
  int tile = blockIdx.x * (blockDim.x >> 5) + wave;
  int row0 = (tile >> 3) * 16, col0 = (tile & 7) * 16;
  v8f acc = {};
  for (int kk = 0; kk < D_; kk += 4) {
    v2f av = ldfragT(X, D_, row0, kk, lane);
    v2f bv = ldfragT(W, D_, col0, kk, lane);
    acc = wmma4(av, bv, acc);
  }
  int n = col0 + (lane & 15);
#pragma unroll
  for (int v = 0; v < 8; ++v) {
    int m = row0 + v + ((lane >> 4) << 3);
    OUT[(size_t)m * D_ + n] = acc[v] + bias[n];
  }
}

// OUT = X @ W   (non-transposed weight, no bias) — used for qk = q @ k_w
__global__ void proj_nn_kernel(const float* __restrict__ X, const float* __restrict__ W,
                               float* __restrict__ OUT) {
  int lane = threadIdx.x & 31;
  int wave = threadIdx.x >> 5;
  int tile = blockIdx.x * (blockDim.x >> 5) + wave;
  int row0 = (tile >> 3) * 16, col0 = (tile & 7) * 16;
  v8f acc = {};
  for (int kk = 0; kk < D_; kk += 4) {
    v2f av = ldfragT(X, D_, row0, kk, lane);
    v2f bv = ldfragN(W, D_, col0, kk, lane);
    acc = wmma4(av, bv, acc);
  }
  int n = col0 + (lane & 15);
#pragma unroll
  for (int v = 0; v < 8; ++v) {
    int m = row0 + v + ((lane >> 4) << 3);
    OUT[(size_t)m * D_ + n] = acc[v];
  }
}

// ---------------------------------------------------------------------------
// LTC scan: one persistent workgroup (8 waves), all 4 HxH weights (256 KB)
// resident in the 320 KB WGP LDS, h (16x128) in LDS. 4 WMMAs per K-chunk.
// ---------------------------------------------------------------------------
#define LTC_LDS_BYTES ((4 * H_ * H_ + B_ * H_) * 4)

__global__ void __launch_bounds__(256) ltc_kernel(
    const float* __restrict__ sensory, const float* __restrict__ taux,
    const float* __restrict__ iw_w,   const float* __restrict__ iw_b,
    const float* __restrict__ imu_w,  const float* __restrict__ imu_b,
    const float* __restrict__ isig_w, const float* __restrict__ isig_b,
    const float* __restrict__ tc_w,   float* __restrict__ ltc_out) {
  extern __shared__ float smem[];
  float* w_iw  = smem;
  float* w_imu = w_iw  + H_ * H_;
  float* w_isg = w_imu + H_ * H_;
  float* w_tch = w_isg + H_ * H_;
  float* hbuf  = w_tch + H_ * H_;   // 16 x 128
  int tid = threadIdx.x;
  for (int i = tid; i < H_ * H_; i += blockDim.x) {
    int hh = i >> 7, j = i & 127;
    w_iw[i]  = iw_w[i];
    w_imu[i] = imu_w[i];
    w_isg[i] = isig_w[i];
    w_tch[i] = tc_w[(size_t)hh * (D_ + H_) + D_ + j];  // tc_h = tc_w[:, D:]
  }
  for (int i = tid; i < B_ * H_; i += blockDim.x) hbuf[i] = 1.0f;  // h0 = ones
  __syncthreads();

  int lane = tid & 31;
  int col0 = (tid >> 5) * 16;        // wave -> 16-col tile
  int n = col0 + (lane & 15);
  for (int t = 0; t < T_; ++t) {
    v8f ai = {}, am = {}, as = {}, at = {};
    for (int kk = 0; kk < H_; kk += 4) {
      v2f av = ldfragT(hbuf, H_, 0, kk, lane);        // rows = 16 batches
      at = wmma4(av, ldfragT(w_tch, H_, col0, kk, lane), at);
      ai = wmma4(av, ldfragT(w_iw,  H_, col0, kk, lane), ai);
      am = wmma4(av, ldfragT(w_imu, H_, col0, kk, lane), am);
      as = wmma4(av, ldfragT(w_isg, H_, col0, kk, lane), as);
    }
    float hn[8];
#pragma unroll
    for (int v = 0; v < 8; ++v) {
      int m = v + ((lane >> 4) << 3);                  // batch
      size_t idx = ((size_t)m * T_ + t) * H_ + n;
      float hv   = hbuf[m * H_ + n];
      float tau  = softplusf_(taux[idx] + at[v]) + 0.1f;
      float inter = (ai[v] + iw_b[n]) * sigmoidf_(am[v] + imu_b[n]) *
                    safe_expf_(as[v] + isig_b[n]);
      float dh = (sensory[idx] + inter - hv) / fmaxf(tau, EPS_);
      hn[v] = hv + DTC_ * dh;
    }
    __syncthreads();                                   // all reads of h done
#pragma unroll
    for (int v = 0; v < 8; ++v) {
      int m = v + ((lane >> 4) << 3);
      hbuf[m * H_ + n] = hn[v];
      ltc_out[((size_t)m * T_ + t) * H_ + n] = hn[v];
    }
    __syncthreads();
  }
}

// ---------------------------------------------------------------------------
// Memory scan: one workgroup per batch; 512x128 memory state resident in LDS
// across all 128 steps. Uses the exact factorization:
//   logit[m] = (q@k_w)·mem[m] + q·k_b ;  mo = (att@mem)@v_w^T + v_b
// ---------------------------------------------------------------------------
#define MEM_LDS_BYTES ((M_ * D_ + M_ + 5 * D_ + 256) * 4 + (256 + 8) * 4)

__global__ void __launch_bounds__(256) mem_kernel(
    const float* __restrict__ memory, const float* __restrict__ ltc_out,
    const float* __restrict__ q_all,  const float* __restrict__ qk_all,
    const float* __restrict__ k_b,
    const float* __restrict__ v_w,    const float* __restrict__ v_b,
    const float* __restrict__ ug_w,   const float* __restrict__ ug_b,
    float* __restrict__ out) {
  extern __shared__ float smem[];
  float* mrows = smem;                  // 512*128
  float* att   = mrows + M_ * D_;       // 512
  float* xt    = att + M_;              // 128
  float* qk    = xt + D_;               // 128
  float* amem  = qk + D_;               // 128
  float* mo    = amem + D_;             // 128
  float* uu    = mo + D_;               // 128
  float* red   = uu + D_;               // 256
  int*   ridx  = (int*)(red + 256);     // 256
  int*   tops  = ridx + 256;            // 8
  int tid = threadIdx.x;
  int b = blockIdx.x;
  const float inv_scale = 1.0f / sqrtf((float)D_);

  for (int i = tid; i < M_ * D_; i += 256) mrows[i] = memory[i];  // broadcast init
  __syncthreads();

  for (int t = 0; t < T_; ++t) {
    size_t r = (size_t)b * T_ + t;
    if (tid < D_) {
      xt[tid]  = ltc_out[r * D_ + tid];
      qk[tid]  = qk_all[r * D_ + tid];
      red[tid] = q_all[r * D_ + tid] * k_b[tid];   // q·k_b partials
    }
    __syncthreads();
    for (int s = 64; s > 0; s >>= 1) {
      if (tid < s) red[tid] += red[tid + s];
      __syncthreads();
    }
    float qkb = red[0];
    __syncthreads();

    // logits: each thread 2 rows of mem
#pragma unroll
    for (int s = 0; s < 2; ++s) {
      int m = tid + s * 256;
      const float* mr = mrows + m * D_;
      float acc = 0.0f;
      for (int d = 0; d < D_; ++d) acc += mr[d] * qk[d];
      att[m] = (acc + qkb) * inv_scale;
    }
    __syncthreads();

    // softmax over 512
    red[tid] = fmaxf(att[tid], att[tid + 256]);
    __syncthreads();
    for (int s = 128; s > 0; s >>= 1) {
      if (tid < s) red[tid] = fmaxf(red[tid], red[tid + s]);
      __syncthreads();
    }
    float mx = red[0];
    __syncthreads();
    float e0 = expf(att[tid] - mx);
    float e1 = expf(att[tid + 256] - mx);
    att[tid] = e0; att[tid + 256] = e1;
    red[tid] = e0 + e1;
    __syncthreads();
    for (int s = 128; s > 0; s >>= 1) {
      if (tid < s) red[tid] += red[tid + s];
      __syncthreads();
    }
    float inv_sum = 1.0f / red[0];
    __syncthreads();
    att[tid] *= inv_sum; att[tid + 256] *= inv_sum;
    __syncthreads();

    // amem = att @ mem : 2 threads per column
    {
      int d = tid & 127, s = tid >> 7;
      float acc = 0.0f;
      for (int m = s * 256; m < s * 256 + 256; ++m) acc += att[m] * mrows[m * D_ + d];
      red[tid] = acc;
      __syncthreads();
      if (tid < D_) amem[tid] = red[tid] + red[tid + 128];
      __syncthreads();
    }

    // mo = amem @ v_w^T + v_b ; write output
    if (tid < D_) {
      const float* wr = v_w + (size_t)tid * D_;
      __builtin_prefetch(wr, 0, 1);
      float acc = v_b[tid];
      for (int d = 0; d < D_; ++d) acc += wr[d] * amem[d];
      mo[tid] = acc;
      out[r * D_ + tid] = acc;
    }
    __syncthreads();

    // u = sigmoid([xt, mo] @ ug_w^T + ug_b)
    if (tid < D_) {
      const float* wr = ug_w + (size_t)tid * (2 * D_);
      float acc = ug_b[tid];
      for (int d = 0; d < D_; ++d) acc += wr[d] * xt[d];
      for (int d = 0; d < D_; ++d) acc += wr[D_ + d] * mo[d];
      uu[tid] = sigmoidf_(acc);
    }
    __syncthreads();

    // top-8 of att (distinct indices, lowest-index tie-break like lax.top_k)
    for (int k8 = 0; k8 < TOPK_; ++k8) {
      float v0 = att[tid], v1 = att[tid + 256];
      float bv = v0; int bi = tid;
      if (v1 > v0) { bv = v1; bi = tid + 256; }
      red[tid] = bv; ridx[tid] = bi;
      __syncthreads();
      for (int s = 128; s > 0; s >>= 1) {
        if (tid < s) {
          float ov = red[tid + s]; int oi = ridx[tid + s];
          if (ov > red[tid] || (ov == red[tid] && oi < ridx[tid])) {
            red[tid] = ov; ridx[tid] = oi;
          }
        }
        __syncthreads();
      }
      if (tid == 0) { int w = ridx[0]; tops[k8] = w; att[w] = -1.0f; }
      __syncthreads();
    }

    // row updates: new = (1-u)*row + u*xt
    {
      int d = tid & 127;
      for (int s = tid >> 7; s < TOPK_; s += 2) {
        float* mr = mrows + tops[s] * D_;
        mr[d] = (1.0f - uu[d]) * mr[d] + uu[d] * xt[d];
      }
    }
    __syncthreads();
  }
}

// ---------------------------------------------------------------------------
extern "C" void kernel_launch(void* const* d_in, const int* in_sizes, int n_in,
                              void* d_out, int out_size, void* d_ws, size_t ws_size,
                              hipStream_t stream) {
  const float* x      = (const float*)d_in[0];
  const float* sw_w   = (const float*)d_in[1];
  const float* sw_b   = (const float*)d_in[2];
  const float* smu_w  = (const float*)d_in[3];
  const float* smu_b  = (const float*)d_in[4];
  const float* ssig_w = (const float*)d_in[5];
  const float* ssig_b = (const float*)d_in[6];
  const float* iw_w   = (const float*)d_in[7];
  const float* iw_b   = (const float*)d_in[8];
  const float* imu_w  = (const float*)d_in[9];
  const float* imu_b  = (const float*)d_in[10];
  const float* isig_w = (const float*)d_in[11];
  const float* isig_b = (const float*)d_in[12];
  const float* tc_w   = (const float*)d_in[13];
  const float* tc_b   = (const float*)d_in[14];
  const float* memory = (const float*)d_in[15];
  const float* q_w    = (const float*)d_in[16];
  const float* q_b    = (const float*)d_in[17];
  const float* k_w    = (const float*)d_in[18];
  const float* k_b    = (const float*)d_in[19];
  const float* v_w    = (const float*)d_in[20];
  const float* v_b    = (const float*)d_in[21];
  const float* ug_w   = (const float*)d_in[22];
  const float* ug_b   = (const float*)d_in[23];

  const size_t NBT = (size_t)B_ * T_ * H_;   // 262144
  float* ws      = (float*)d_ws;
  float* sensory = ws;
  float* taux    = sensory + NBT;
  float* ltc_out = taux + NBT;
  float* q_all   = ltc_out + NBT;
  float* qk_all  = q_all + NBT;

  hipFuncSetAttribute(reinterpret_cast<const void*>(ltc_kernel),
                      hipFuncAttributeMaxDynamicSharedMemorySize, LTC_LDS_BYTES);
  hipFuncSetAttribute(reinterpret_cast<const void*>(mem_kernel),
                      hipFuncAttributeMaxDynamicSharedMemorySize, MEM_LDS_BYTES);

  // Stage A: 1024 wave-tiles, 4 waves/block
  stage_a_kernel<<<256, 128, 0, stream>>>(x, sw_w, sw_b, smu_w, smu_b,
                                          ssig_w, ssig_b, tc_w, tc_b,
                                          sensory, taux);
  // LTC sequential scan, LDS-resident weights
  ltc_kernel<<<1, 256, LTC_LDS_BYTES, stream>>>(sensory, taux, iw_w, iw_b,
                                                imu_w, imu_b, isig_w, isig_b,
                                                tc_w, ltc_out);
  // q_all = ltc_out @ q_w^T + q_b ; qk_all = q_all @ k_w
  proj_nt_kernel<<<256, 128, 0, stream>>>(ltc_out, q_w, q_b, q_all);
  proj_nn_kernel<<<256, 128, 0, stream>>>(q_all, k_w, qk_all);
  // Memory scan: one workgroup per batch, memory state in LDS
  mem_kernel<<<B_, 256, MEM_LDS_BYTES, stream>>>(memory, ltc_out, q_all, qk_all,
                                                 k_b, v_w, v_b, ug_w, ug_b,
                                                 (float*)d_out);
}